// DistGATConv_46720654246115
// MI455X (gfx1250) — compile-verified
//
#include <hip/hip_runtime.h>

// ---------------------------------------------------------------------------
// DistGATConv for MI455X (gfx1250, wave32).
// Phases:
//   1) h = x @ fc_w^T via V_WMMA_F32_16X16X4_F32 (exact fp32 GEMM)
//   2) el[n,h] = <h[n,h,:], attn_l[h,:]> ; er via dst_node_ids gather
//   3) per-edge: e = leaky_relu(el[src]+er[dst]); exp; atomic denominator
//   4) invert denominators (divide once per dst, not per edge)
//   5) wave-per-edge weighted scatter-add into out (L2-resident atomics)
// ---------------------------------------------------------------------------

typedef __attribute__((ext_vector_type(2))) float v2f;
typedef __attribute__((ext_vector_type(4))) float v4f;
typedef __attribute__((ext_vector_type(8))) float v8f;

#define N_SRC 100000
#define N_DST 50000
#define N_EDGE 1000000
#define IN_F 256
#define NH 4
#define ND 32
#define HD 128            // NH*ND
#define NEG_SLOPE 0.2f

// ---------------------------------------------------------------------------
// Kernel 1: FC GEMM  h[N_SRC,128] = x[N_SRC,256] @ w[128,256]^T
// Each wave computes one 16-row x 128-col strip with 8 f32 accumulators.
// A frag (16x4 f32): lane l (r=l&15, hi=l>>4): v0=x[m0+r][k+2hi], v1=..k+2hi+1
// B frag (4x16 f32): lane l:                   v0=w[n0+r][k+2hi], v1=..k+2hi+1
// D tile layout: acc[j][v] = h[m0 + v + 8*hi][j*16 + r]
// ---------------------------------------------------------------------------
__global__ __launch_bounds__(256) void gat_fc_wmma(
    const float* __restrict__ x, const float* __restrict__ w,
    float* __restrict__ hbuf)
{
    const int wave  = threadIdx.x >> 5;
    const int lane  = threadIdx.x & 31;
    const int strip = blockIdx.x * 8 + wave;          // 16-row strip index
    if (strip >= (N_SRC / 16)) return;                // wave-uniform exit
    const int m0 = strip * 16;
    const int r  = lane & 15;
    const int hi = lane >> 4;

    const float* __restrict__ xrow  = x + (size_t)(m0 + r) * IN_F + 2 * hi;
    const float* __restrict__ wbase = w + (size_t)r * IN_F + 2 * hi;

    v8f acc[8] = {};

#pragma unroll 2
    for (int k = 0; k < IN_F; k += 4) {
        v2f a = *(const v2f*)(xrow + k);
#pragma unroll
        for (int j = 0; j < 8; ++j) {
            v2f b = *(const v2f*)(wbase + (size_t)j * 16 * IN_F + k);
            acc[j] = __builtin_amdgcn_wmma_f32_16x16x4_f32(
                false, a, false, b, (short)0, acc[j], false, false);
        }
    }

#pragma unroll
    for (int j = 0; j < 8; ++j) {
#pragma unroll
        for (int v = 0; v < 8; ++v) {
            hbuf[(size_t)(m0 + v + 8 * hi) * HD + j * 16 + r] = acc[j][v];
        }
    }
}

// ---------------------------------------------------------------------------
// Kernel 2a: el[n,h] = sum_d h[n,h,d]*attn_l[h,d]      (one thread per (n,h))
// ---------------------------------------------------------------------------
__global__ void gat_el(const float* __restrict__ hbuf,
                       const float* __restrict__ attn,
                       float* __restrict__ el)
{
    int idx = blockIdx.x * blockDim.x + threadIdx.x;
    if (idx >= N_SRC * NH) return;
    int node = idx >> 2, hh = idx & 3;
    const float* hp = hbuf + (size_t)node * HD + hh * ND;
    const float* ap = attn + hh * ND;
    float s = 0.f;
#pragma unroll
    for (int d = 0; d < ND; d += 4) {
        v4f hv = *(const v4f*)(hp + d);
        v4f av = *(const v4f*)(ap + d);
        s += hv.x * av.x + hv.y * av.y + hv.z * av.z + hv.w * av.w;
    }
    el[idx] = s;
}

// ---------------------------------------------------------------------------
// Kernel 2b: er[n,h] = sum_d h[dst_node_ids[n],h,d]*attn_r[h,d]
// ---------------------------------------------------------------------------
__global__ void gat_er(const float* __restrict__ hbuf,
                       const float* __restrict__ attn,
                       const int* __restrict__ ids,
                       float* __restrict__ er)
{
    int idx = blockIdx.x * blockDim.x + threadIdx.x;
    if (idx >= N_DST * NH) return;
    int node = ids[idx >> 2];
    int hh = idx & 3;
    const float* hp = hbuf + (size_t)node * HD + hh * ND;
    const float* ap = attn + hh * ND;
    float s = 0.f;
#pragma unroll
    for (int d = 0; d < ND; d += 4) {
        v4f hv = *(const v4f*)(hp + d);
        v4f av = *(const v4f*)(ap + d);
        s += hv.x * av.x + hv.y * av.y + hv.z * av.z + hv.w * av.w;
    }
    er[idx] = s;
}

// ---------------------------------------------------------------------------
// Kernel 3: zero a buffer (harness poisons d_out / d_ws with 0xAA)
// ---------------------------------------------------------------------------
__global__ void gat_zero(float* __restrict__ p, int n)
{
    int i = blockIdx.x * blockDim.x + threadIdx.x;
    if (i < n) p[i] = 0.f;
}

// ---------------------------------------------------------------------------
// Kernel 4: per-edge exp(leaky_relu(el[src]+er[dst])) + atomic denominator
// ---------------------------------------------------------------------------
__global__ void gat_edge(const int* __restrict__ src_idx,
                         const int* __restrict__ dst_idx,
                         const float* __restrict__ el,
                         const float* __restrict__ er,
                         float* __restrict__ exp_e,
                         float* __restrict__ sum_exp)
{
    int e = blockIdx.x * blockDim.x + threadIdx.x;
    if (e >= N_EDGE) return;
    int s = src_idx[e], d = dst_idx[e];
    v4f l = *(const v4f*)(el + 4 * (size_t)s);
    v4f r = *(const v4f*)(er + 4 * (size_t)d);
    v4f ex;
#pragma unroll
    for (int hh = 0; hh < NH; ++hh) {
        float v = l[hh] + r[hh];
        v = (v >= 0.f) ? v : NEG_SLOPE * v;
        float xv = expf(v);
        ex[hh] = xv;
        atomicAdd(&sum_exp[4 * (size_t)d + hh], xv);
    }
    *(v4f*)(exp_e + 4 * (size_t)e) = ex;
}

// ---------------------------------------------------------------------------
// Kernel 5: invert denominators in place
// ---------------------------------------------------------------------------
__global__ void gat_invert(float* __restrict__ p, int n)
{
    int i = blockIdx.x * blockDim.x + threadIdx.x;
    if (i < n) p[i] = 1.0f / p[i];
}

// ---------------------------------------------------------------------------
// Kernel 6: wave-per-edge weighted scatter.
// Lane l handles features f = hh*32 + l  (hh = 0..3): reads of h[src] and
// atomics into out[dst] are each one contiguous 128B transaction per wave.
// h (51 MB) and out (26 MB) are L2-resident (192 MB L2) -> atomics stay in L2.
// ---------------------------------------------------------------------------
__global__ __launch_bounds__(256) void gat_scatter(
    const int* __restrict__ src_idx, const int* __restrict__ dst_idx,
    const float* __restrict__ exp_e, const float* __restrict__ inv_sum,
    const float* __restrict__ hbuf, float* __restrict__ out)
{
    int wave = threadIdx.x >> 5;
    int lane = threadIdx.x & 31;
    int e = blockIdx.x * 8 + wave;
    if (e >= N_EDGE) return;
    int s = src_idx[e], d = dst_idx[e];
    const float* __restrict__ hs = hbuf + (size_t)s * HD;
    float* __restrict__ op = out + (size_t)d * HD;
#pragma unroll
    for (int hh = 0; hh < NH; ++hh) {
        float att = exp_e[4 * (size_t)e + hh] * inv_sum[4 * (size_t)d + hh];
        int f = hh * ND + lane;
        atomicAdd(&op[f], att * hs[f]);
    }
}

// ---------------------------------------------------------------------------
extern "C" void kernel_launch(void* const* d_in, const int* in_sizes, int n_in,
                              void* d_out, int out_size, void* d_ws, size_t ws_size,
                              hipStream_t stream)
{
    const float* x           = (const float*)d_in[0];   // [N_SRC, 256]
    const float* fc_w        = (const float*)d_in[1];   // [128, 256]
    const float* attn_l      = (const float*)d_in[2];   // [1,4,32] -> 128
    const float* attn_r      = (const float*)d_in[3];   // [1,4,32] -> 128
    const int*   src_idx     = (const int*)d_in[4];     // [E]
    const int*   dst_idx     = (const int*)d_in[5];     // [E]
    const int*   dst_node_id = (const int*)d_in[6];     // [N_DST]
    float*       out         = (float*)d_out;           // [N_DST, 128]

    // Workspace layout (floats): ~70.5 MB total
    float* ws    = (float*)d_ws;
    float* hbuf  = ws;                                   // N_SRC*128
    float* el    = hbuf + (size_t)N_SRC * HD;            // N_SRC*4
    float* er    = el + (size_t)N_SRC * NH;              // N_DST*4
    float* expb  = er + (size_t)N_DST * NH;              // E*4
    float* sume  = expb + (size_t)N_EDGE * NH;           // N_DST*4

    const int B = 256;

    // zero accumulation targets (harness poisons them with 0xAA)
    gat_zero<<<(N_DST * HD + B - 1) / B, B, 0, stream>>>(out, N_DST * HD);
    gat_zero<<<(N_DST * NH + B - 1) / B, B, 0, stream>>>(sume, N_DST * NH);

    // 1) fp32 WMMA GEMM: 6250 strips, 8 waves/block
    gat_fc_wmma<<<(N_SRC / 16 + 7) / 8, B, 0, stream>>>(x, fc_w, hbuf);

    // 2) per-node attention scalars
    gat_el<<<(N_SRC * NH + B - 1) / B, B, 0, stream>>>(hbuf, attn_l, el);
    gat_er<<<(N_DST * NH + B - 1) / B, B, 0, stream>>>(hbuf, attn_r, dst_node_id, er);

    // 3) edge softmax numerators + denominators
    gat_edge<<<(N_EDGE + B - 1) / B, B, 0, stream>>>(src_idx, dst_idx, el, er, expb, sume);

    // 4) reciprocal of denominators
    gat_invert<<<(N_DST * NH + B - 1) / B, B, 0, stream>>>(sume, N_DST * NH);

    // 5) weighted scatter (one wave per edge)
    gat_scatter<<<(N_EDGE + 7) / 8, B, 0, stream>>>(src_idx, dst_idx, expb, sume, hbuf, out);
}